// EdgeModule_8074538516997
// MI455X (gfx1250) — compile-verified
//
#include <hip/hip_runtime.h>

// Shapes fixed by the reference: B=2, L=1024, H=256, E=64, N_TYPES=5.
#define BB   2
#define LL   1024
#define HH   256
#define EE   64
#define NT   5

typedef __attribute__((ext_vector_type(2))) float v2f;
typedef __attribute__((ext_vector_type(8))) float v8f;
typedef __attribute__((ext_vector_type(4))) float f32x4;

// ---------------------------------------------------------------------------
// Kernel 1: per-type fused matrices  CW[s][t] = T[s][t]^T @ Whalf[s]  (256x64)
// D = A x B with M=d (16 tiles), N=e (4 tiles), K=m (256, step 4) using
// V_WMMA_F32_16X16X4_F32. One wave (32 lanes, EXEC all ones) per 16x16 tile.
// grid.x = 2 sides * 5 types * 16 dtiles * 4 etiles = 640 waves.
// ---------------------------------------------------------------------------
__global__ __launch_bounds__(32) void edge_precompute_wmma(
    const float* __restrict__ lt,   // [5,256,256]  (t,m,d)
    const float* __restrict__ rt,   // [5,256,256]
    const float* __restrict__ W,    // [512,64]     (m,e)
    float* __restrict__ CW)         // [2,5,256,64] (s,t,d,e)
{
    int id    = blockIdx.x;
    int etile = id & 3;  id >>= 2;
    int dtile = id & 15; id >>= 4;
    int t     = id % NT;
    int s     = id / NT;

    const float* T  = (s == 0 ? lt : rt) + t * (HH * HH);
    const float* Wh = W + s * (HH * EE);

    int lane = threadIdx.x;
    int half = lane >> 4;     // 0: lanes 0-15 hold K,K+1 ; 1: lanes 16-31 hold K+2,K+3
    int lidx = lane & 15;
    int d    = dtile * 16 + lidx;   // M index for A lanes
    int e    = etile * 16 + lidx;   // N index for B lanes

    v8f acc = {0.f, 0.f, 0.f, 0.f, 0.f, 0.f, 0.f, 0.f};

    #pragma unroll 4
    for (int k0 = 0; k0 < HH; k0 += 4) {
        int m = k0 + 2 * half;
        v2f a, b;
        // A(Mrow=d, K=m) = T[m][d]  (this *is* the transpose: row of T^T)
        a[0] = T[(m + 0) * HH + d];
        a[1] = T[(m + 1) * HH + d];
        // B(K=m, Ncol=e) = Wh[m][e]
        b[0] = Wh[(m + 0) * EE + e];
        b[1] = Wh[(m + 1) * EE + e];
        acc = __builtin_amdgcn_wmma_f32_16x16x4_f32(
            /*neg_a=*/false, a, /*neg_b=*/false, b,
            /*c_mod=*/(short)0, acc, /*reuse_a=*/false, /*reuse_b=*/false);
    }

    // D layout: VGPR r -> (M = r + 8*half, N = lidx)
    float* outp = CW + (((size_t)(s * NT + t)) * HH + dtile * 16) * EE + etile * 16;
    #pragma unroll
    for (int r = 0; r < 8; ++r) {
        outp[(r + 8 * half) * EE + lidx] = acc[r];
    }
}

// ---------------------------------------------------------------------------
// Kernel 2: per-token projection with type gather.
//   lp[tok][e] = x[tok] . CW[0][type][:,e] + bias[e]   (bias folded here)
//   rp[tok][e] = x[tok] . CW[1][type][:,e]
// One block (128 threads) per token; x staged in LDS; CW columns are
// coalesced across the 64 e-threads and L2-resident (640 KB total).
// ---------------------------------------------------------------------------
__global__ __launch_bounds__(128) void edge_project(
    const float* __restrict__ x,      // [B*L, 256]
    const int*   __restrict__ types,  // [B*L]
    const float* __restrict__ CW,     // [2,5,256,64]
    const float* __restrict__ bias,   // [64]
    float* __restrict__ lp,           // [B*L, 64]
    float* __restrict__ rp)           // [B*L, 64]
{
    __shared__ float xs[HH];
    int tok = blockIdx.x;
    int tid = threadIdx.x;

    xs[tid]       = x[(size_t)tok * HH + tid];
    xs[tid + 128] = x[(size_t)tok * HH + tid + 128];
    __syncthreads();

    int t = types[tok];
    int s = tid >> 6;          // 0 = left, 1 = right
    int e = tid & 63;

    const float* M = CW + ((size_t)(s * NT + t)) * (HH * EE) + e;
    float sum = 0.f;
    #pragma unroll 8
    for (int d = 0; d < HH; ++d) sum = fmaf(xs[d], M[(size_t)d * EE], sum);

    if (s == 0) {
        lp[(size_t)tok * EE + e] = sum + bias[e];
    } else {
        rp[(size_t)tok * EE + e] = sum;
    }
}

// ---------------------------------------------------------------------------
// Kernel 3: out[b,i,j,:] = lp[b,i,:] + rp[b,j,:]
// Pure streaming-store kernel: 512 MiB of output at HBM rate. lp/rp are
// 1 MiB total -> L2 hits. One block per (b,i) row; float4 lanes; j chunked;
// non-temporal b128 stores keep the stream out of the caches.
// ---------------------------------------------------------------------------
__global__ __launch_bounds__(256) void edge_pairsum(
    const f32x4* __restrict__ lp4,   // [B*L, 16] float4
    const f32x4* __restrict__ rp4,   // [B*L, 16] float4
    f32x4* __restrict__ out)         // [B*L, L, 16] float4
{
    int row = blockIdx.x;            // b*L + i
    int b   = row >> 10;             // L = 1024
    int tid = threadIdx.x;
    int e4  = tid & 15;              // float4 column within E=64
    int j0  = tid >> 4;              // 0..15

    f32x4 l = lp4[(size_t)row * 16 + e4];
    const f32x4* rrow = rp4 + (size_t)b * (LL * 16);
    f32x4*       orow = out + (size_t)row * (LL * 16);

    #pragma unroll 4
    for (int jj = 0; jj < 64; ++jj) {
        int j = jj * 16 + j0;
        f32x4 r = rrow[(size_t)j * 16 + e4];
        f32x4 v = l + r;
        __builtin_nontemporal_store(v, &orow[(size_t)j * 16 + e4]);
    }
}

// ---------------------------------------------------------------------------
extern "C" void kernel_launch(void* const* d_in, const int* in_sizes, int n_in,
                              void* d_out, int out_size, void* d_ws, size_t ws_size,
                              hipStream_t stream) {
    const float* token_embs  = (const float*)d_in[0];  // [2,1024,256]
    const int*   token_types = (const int*)  d_in[1];  // [2,1024]
    const float* left_tf     = (const float*)d_in[2];  // [5,256,256]
    const float* right_tf    = (const float*)d_in[3];  // [5,256,256]
    const float* W           = (const float*)d_in[4];  // [512,64]
    const float* bias        = (const float*)d_in[5];  // [64]
    float*       out         = (float*)d_out;          // [2,1024,1024,64]

    // Workspace carve-up (floats): CW 163840 | lp 131072 | rp 131072  (~1.7 MB)
    float* CW = (float*)d_ws;
    float* lp = CW + (size_t)2 * NT * HH * EE;
    float* rp = lp + (size_t)BB * LL * EE;

    // 1) fused per-type matrices via f32 WMMA (2*5*16*4 tiles, 1 wave each)
    edge_precompute_wmma<<<2 * NT * 16 * 4, 32, 0, stream>>>(left_tf, right_tf, W, CW);

    // 2) per-token gather + projection (bias folded into lp)
    edge_project<<<BB * LL, 128, 0, stream>>>(token_embs, token_types, CW, bias, lp, rp);

    // 3) broadcast pair-sum, 512 MiB NT streaming stores (the roofline term)
    edge_pairsum<<<BB * LL, 256, 0, stream>>>((const f32x4*)lp, (const f32x4*)rp,
                                              (f32x4*)out);
}